// Decoder_76227079569945
// MI455X (gfx1250) — compile-verified
//
#include <hip/hip_runtime.h>
#include <hip/hip_bf16.h>
#include <math.h>

#define H 1024
#define V 50257

typedef __attribute__((ext_vector_type(2))) float v2f;
typedef __attribute__((ext_vector_type(8))) float v8f;

// ---------------------------------------------------------------------------
// Kernel 1: GRU cell. 64 workgroups x 256 threads; WG b handles h rows
// [b*16, b*16+16). 8 waves * 12 row-dot-products (6 per output element).
// ---------------------------------------------------------------------------
__global__ void gru_kernel(const int* __restrict__ ids,
                           const float* __restrict__ hidden,
                           const float* __restrict__ emb,
                           const float* __restrict__ W_ih,
                           const float* __restrict__ W_hh,
                           const float* __restrict__ b_ih,
                           const float* __restrict__ b_hh,
                           float* __restrict__ h_ws,
                           float* __restrict__ out_hidden) {
  __shared__ float xs[H];
  __shared__ float hsv[H];
  __shared__ float res[96];

  const int tid = threadIdx.x;
  const int id = ids[0];
  for (int i = tid; i < H; i += 256) {
    float xv = emb[(size_t)id * H + i];
    xs[i] = xv > 0.0f ? xv : 0.0f;   // relu(embedding)
    hsv[i] = hidden[i];
  }
  __syncthreads();

  const int lane = tid & 31;
  const int wave = tid >> 5;

  // task t = mat*48 + gate*16 + jj  (mat: 0=W_ih&x, 1=W_hh&h; gate: 0=r,1=z,2=n)
  for (int t = wave * 12; t < wave * 12 + 12; ++t) {
    const int mat = t / 48;
    const int sub = t % 48;
    const int gate = sub / 16;
    const int jj = sub % 16;
    const int row = gate * H + blockIdx.x * 16 + jj;
    const float* Wrow = (mat == 0 ? W_ih : W_hh) + (size_t)row * H;
    const float* vec = (mat == 0 ? xs : hsv);

    float acc = 0.0f;
#pragma unroll
    for (int k = lane * 4; k < H; k += 32 * 4) {
      float4 w = *(const float4*)(Wrow + k);
      float4 x = *(const float4*)(vec + k);
      acc += w.x * x.x + w.y * x.y + w.z * x.z + w.w * x.w;
    }
#pragma unroll
    for (int off = 16; off > 0; off >>= 1)
      acc += __shfl_xor(acc, off, 32);
    if (lane == 0) res[t] = acc;
  }
  __syncthreads();

  if (tid < 16) {
    const int j = blockIdx.x * 16 + tid;
    const float i_r = res[0 * 16 + tid] + b_ih[j];
    const float i_z = res[1 * 16 + tid] + b_ih[H + j];
    const float i_n = res[2 * 16 + tid] + b_ih[2 * H + j];
    const float h_r = res[48 + 0 * 16 + tid] + b_hh[j];
    const float h_z = res[48 + 1 * 16 + tid] + b_hh[H + j];
    const float h_n = res[48 + 2 * 16 + tid] + b_hh[2 * H + j];
    const float r = 1.0f / (1.0f + expf(-(i_r + h_r)));
    const float z = 1.0f / (1.0f + expf(-(i_z + h_z)));
    const float n = tanhf(i_n + r * h_n);
    const float hn = (1.0f - z) * n + z * hsv[j];
    h_ws[j] = hn;        // feeds logits kernel
    out_hidden[j] = hn;  // hidden_out part of d_out
  }
}

// ---------------------------------------------------------------------------
// Kernel 2: logits = W_out @ h + b_out via V_WMMA_F32_16X16X4_F32.
// Each wave owns one 16-row tile of W_out; B = h replicated across N columns.
// A layout (16x4 f32): lane m (0-15): K=k0,k0+1 ; lane m+16: K=k0+2,k0+3 ->
// one contiguous b64 per lane per K-step. 1024/4 = 256 WMMAs per tile.
// ---------------------------------------------------------------------------
__global__ void logits_wmma(const float* __restrict__ Wout,
                            const float* __restrict__ b_out,
                            const float* __restrict__ hvec,
                            float* __restrict__ logits) {
  __shared__ float hs[H];
  const int tid = threadIdx.x;
  for (int i = tid; i < H; i += 128) hs[i] = hvec[i];
  __syncthreads();

  const int lane = tid & 31;
  const int wave = tid >> 5;
  const int m0 = (blockIdx.x * 4 + wave) * 16;
  if (m0 >= V) return;  // wave-uniform: EXEC stays all-1s for live waves

  const int mrow = m0 + (lane & 15);
  const int rowc = mrow < V ? mrow : (V - 1);  // clamp loads in last tile
  const float* wrow = Wout + (size_t)rowc * H;
  const int khalf = (lane < 16) ? 0 : 2;

  v8f c = {};
#pragma unroll 8
  for (int k0 = 0; k0 < H; k0 += 4) {
    v2f a = *(const v2f*)(wrow + k0 + khalf);          // global b64, line fully reused
    v2f b = *(const v2f*)(&hs[k0 + khalf]);            // ds b64
    c = __builtin_amdgcn_wmma_f32_16x16x4_f32(
        /*neg_a=*/false, a, /*neg_b=*/false, b,
        /*c_mod=*/(short)0, c, /*reuse_a=*/false, /*reuse_b=*/false);
  }

  // D layout: lane 0 holds rows m0..m0+7 (N=0), lane 16 holds rows m0+8..m0+15.
  if ((lane & 15) == 0) {
    const int base = m0 + ((lane == 16) ? 8 : 0);
#pragma unroll
    for (int r = 0; r < 8; ++r) {
      const int row = base + r;
      if (row < V) logits[row] = c[r] + b_out[row];
    }
  }
}

// ---------------------------------------------------------------------------
// Kernel 3: single-WG log-softmax stats: stats[0]=max, stats[1]=log(sum exp).
// ---------------------------------------------------------------------------
__global__ void softmax_reduce(const float* __restrict__ logits,
                               float* __restrict__ stats) {
  __shared__ float red[1024];
  const int tid = threadIdx.x;

  float m = -INFINITY;
  for (int i = tid; i < V; i += 1024) m = fmaxf(m, logits[i]);
  red[tid] = m;
  __syncthreads();
  for (int s = 512; s > 0; s >>= 1) {
    if (tid < s) red[tid] = fmaxf(red[tid], red[tid + s]);
    __syncthreads();
  }
  const float M = red[0];
  __syncthreads();

  float ssum = 0.0f;
  for (int i = tid; i < V; i += 1024) ssum += expf(logits[i] - M);
  red[tid] = ssum;
  __syncthreads();
  for (int s = 512; s > 0; s >>= 1) {
    if (tid < s) red[tid] += red[tid + s];
    __syncthreads();
  }
  if (tid == 0) {
    stats[0] = M;
    stats[1] = logf(red[0]);
  }
}

// ---------------------------------------------------------------------------
// Kernel 4: out[v] = logits[v] - max - log(sum)
// ---------------------------------------------------------------------------
__global__ void finalize(const float* __restrict__ logits,
                         const float* __restrict__ stats,
                         float* __restrict__ out) {
  const int v = blockIdx.x * blockDim.x + threadIdx.x;
  if (v < V) out[v] = logits[v] - stats[0] - stats[1];
}

// ---------------------------------------------------------------------------
// Launch. Inputs (setup_inputs order):
// 0 input_ids(i32) 1 hidden 2 emb 3 W_ih 4 W_hh 5 b_ih 6 b_hh 7 W_out 8 b_out
// d_out: [0..V) logp, [V..V+H) hidden_out.
// ws (floats): [0..H) h_new | [H..H+V) logits | [H+V+ pad] stats(2)
// ---------------------------------------------------------------------------
extern "C" void kernel_launch(void* const* d_in, const int* in_sizes, int n_in,
                              void* d_out, int out_size, void* d_ws, size_t ws_size,
                              hipStream_t stream) {
  const int* ids = (const int*)d_in[0];
  const float* hidden = (const float*)d_in[1];
  const float* emb = (const float*)d_in[2];
  const float* W_ih = (const float*)d_in[3];
  const float* W_hh = (const float*)d_in[4];
  const float* b_ih = (const float*)d_in[5];
  const float* b_hh = (const float*)d_in[6];
  const float* W_out = (const float*)d_in[7];
  const float* b_out = (const float*)d_in[8];
  float* out = (float*)d_out;

  float* ws = (float*)d_ws;
  float* h_ws = ws;                 // H floats
  float* logits = ws + H;           // V floats
  float* stats = ws + H + V + 15;   // 2 floats (padded)

  gru_kernel<<<H / 16, 256, 0, stream>>>(ids, hidden, emb, W_ih, W_hh,
                                         b_ih, b_hh, h_ws, out + V);

  const int tiles = (V + 15) / 16;        // 3142 waves
  const int wgs = (tiles + 3) / 4;        // 4 waves (128 thr) per WG
  logits_wmma<<<wgs, 128, 0, stream>>>(W_out, b_out, h_ws, logits);

  softmax_reduce<<<1, 1024, 0, stream>>>(logits, stats);

  finalize<<<(V + 255) / 256, 256, 0, stream>>>(logits, stats, out);
}